// LevelHolderGate_50319836840001
// MI455X (gfx1250) — compile-verified
//
#include <hip/hip_runtime.h>

typedef float v2f __attribute__((ext_vector_type(2)));
typedef float v8f __attribute__((ext_vector_type(8)));

#ifndef __has_builtin
#define __has_builtin(x) 0
#endif
#if __has_builtin(__builtin_amdgcn_wmma_f32_16x16x4_f32)
#define HAVE_WMMA_F32X4 1
#else
#define HAVE_WMMA_F32X4 0
#endif

#define HW 512
#define TS 32                 // output tile (32x32)
#define EXT 37                // extended tile (TS + 2 lo-halo + 3 hi-halo)
#define EPIX (EXT * EXT)      // 1369 extended pixels
#define NCHUNK ((EPIX + 15) / 16)   // 86 WMMA chunks
#define EPAD (NCHUNK * 16)    // 1376: sD padded so chunk stores need no guard

// ls_c[i] = log(i+1) - mean(log 1..6); INVS = 1 / sum(ls_c^2)
#define LS0 (-1.09654187f)
#define LS1 (-0.40339470f)
#define LS2 ( 0.00207042f)
#define LS3 ( 0.28975248f)
#define LS4 ( 0.51289606f)
#define LS5 ( 0.69521758f)
#define INVS ( 0.45548097f)

// ---------------- phase 1: d = relu(BN(x @ w_down + b_down)) -> LDS ----------
// V_WMMA_F32_16X16X4_F32: D[16x16] = A[16x4]*B[4x16] + C, exact f32 math.
//   A rows = 16 extended-tile pixels, K = input channel (K3 = zero pad)
//   B[k][n] = w_down[k*3+n] (n<3)      -> D columns = d-channels
// CHK=false: whole extended tile (incl. chunk padding) is inside the image,
// so loads and stores are completely branch-free.
template <bool CHK>
__device__ __forceinline__ void down_proj(
    const float* __restrict__ ximg, const float* __restrict__ w_down,
    const float* __restrict__ b_down, const float* __restrict__ g1,
    const float* __restrict__ b1, const float* __restrict__ m1,
    const float* __restrict__ v1, float* __restrict__ sD, int x0, int y0,
    int lane, int wave) {
#if HAVE_WMMA_F32X4
  const int nc = lane & 15;
  v2f bop;
  bop.x = 0.f;
  bop.y = 0.f;
  if (nc < 3) {
    if (lane < 16) {  // VGPR0 = row K0, VGPR1 = row K1
      bop.x = w_down[nc];
      bop.y = w_down[3 + nc];
    } else {          // VGPR0 = row K2, VGPR1 = row K3 (=0)
      bop.x = w_down[6 + nc];
    }
  }
  float bd = 0.f, sc1 = 0.f, bb1 = 0.f, mm1 = 0.f;
  if (nc < 3) {
    bd = b_down[nc];
    sc1 = g1[nc] * rsqrtf(v1[nc] + 1e-3f);
    bb1 = b1[nc];
    mm1 = m1[nc];
  }
  for (int cp = wave; cp < NCHUNK; cp += 8) {  // uniform per wave
    const int pA = cp * 16 + nc;               // this lane's A-row pixel
    const int eyA = pA / EXT, exA = pA - eyA * EXT;
    const int gy = y0 - 2 + eyA, gx = x0 - 2 + exA;
    v2f aop;
    aop.x = 0.f;
    aop.y = 0.f;
    if (!CHK || ((unsigned)gy < HW && (unsigned)gx < HW)) {
      const float* px = ximg + ((size_t)gy * HW + gx) * 3;
      if (lane < 16) {  // supply K0,K1 = channels 0,1
        aop.x = px[0];
        aop.y = px[1];
      } else {          // supply K2 = channel 2
        aop.x = px[2];
      }
    }
    v8f acc = {0.f, 0.f, 0.f, 0.f, 0.f, 0.f, 0.f, 0.f};
    acc = __builtin_amdgcn_wmma_f32_16x16x4_f32(false, aop, false, bop,
                                                (short)0, acc, false, false);
    // D layout: lane L<16 -> column L, rows 0..7 (VGPR j = row j);
    //           lane L>=16 -> column L-16, rows 8..15.
    if (nc < 3) {
      const int p0 = cp * 16 + ((lane < 16) ? 0 : 8);
      const int a0 = p0 * 3 + nc;
#pragma unroll
      for (int j = 0; j < 8; ++j) {
        float v = acc[j] + bd;
        v = fmaf(v - mm1, sc1, bb1);
        v = fmaxf(v, 0.f);
        if (CHK) {  // conv-input zero padding outside the image
          const int p = p0 + j;
          const int ey = p / EXT, ex = p - ey * EXT;
          const int gy2 = y0 - 2 + ey, gx2 = x0 - 2 + ex;
          if (!((unsigned)gy2 < HW && (unsigned)gx2 < HW)) v = 0.f;
        }
        sD[a0 + j * 3] = v;
      }
    }
  }
#else
  const int tid = wave * 32 + lane;
  float sc1c[3], bb1c[3], mm1c[3], bdc[3];
#pragma unroll
  for (int c = 0; c < 3; ++c) {
    sc1c[c] = g1[c] * rsqrtf(v1[c] + 1e-3f);
    bb1c[c] = b1[c];
    mm1c[c] = m1[c];
    bdc[c] = b_down[c];
  }
  for (int p = tid; p < EPIX; p += 256) {
    const int ey = p / EXT, ex = p - ey * EXT;
    const int gy = y0 - 2 + ey, gx = x0 - 2 + ex;
    float dv[3] = {0.f, 0.f, 0.f};
    if (!CHK || ((unsigned)gy < HW && (unsigned)gx < HW)) {
      const float* px = ximg + ((size_t)gy * HW + gx) * 3;
      const float c0 = px[0], c1 = px[1], c2 = px[2];
#pragma unroll
      for (int c = 0; c < 3; ++c) {
        float v = c0 * w_down[c] + c1 * w_down[3 + c] + c2 * w_down[6 + c] + bdc[c];
        v = fmaf(v - mm1c[c], sc1c[c], bb1c[c]);
        dv[c] = fmaxf(v, 0.f);
      }
    }
#pragma unroll
    for (int c = 0; c < 3; ++c) sD[p * 3 + c] = dv[c];
  }
#endif
}

// ---------------- phase 2 helper: one Gaussian scale, 4 output rows ----------
// Register-blocked over 4 consecutive output rows: the K+3 input rows are each
// read once from LDS and reused by up to 4 outputs (LDS traffic ~2.4x lower
// than the direct form). Weights come from global via uniform indices ->
// s_load into SGPRs (constant cache), zero LDS/VALU cost per use.
// LO = lax SAME pad_lo = (K-1)/2.
template <int K, int LO>
__device__ __forceinline__ void accum_scale4(const float* __restrict__ sD,
                                             const float* __restrict__ gk,
                                             int yl0, int xl, float lsc,
                                             float (&h)[4][3]) {
  float w[K * K];
#pragma unroll
  for (int i = 0; i < K * K; ++i) w[i] = gk[i];
  float acc[4][3];
#pragma unroll
  for (int q = 0; q < 4; ++q)
#pragma unroll
    for (int cc = 0; cc < 3; ++cc) acc[q][cc] = 0.f;

  const int ey0 = yl0 + 2 - LO;
  const int ex0 = xl + 2 - LO;
#pragma unroll
  for (int rr = 0; rr < K + 3; ++rr) {  // union of input rows for 4 outputs
    const float* drow = &sD[((ey0 + rr) * EXT + ex0) * 3];
#pragma unroll
    for (int c = 0; c < K; ++c) {
      const float d0 = drow[c * 3 + 0];
      const float d1 = drow[c * 3 + 1];
      const float d2 = drow[c * 3 + 2];
#pragma unroll
      for (int q = 0; q < 4; ++q) {
        const int r = rr - q;  // kernel row feeding output q from input row rr
        if (r >= 0 && r < K) {
          const float wt = w[r * K + c];
          acc[q][0] = fmaf(wt, d0, acc[q][0]);
          acc[q][1] = fmaf(wt, d1, acc[q][1]);
          acc[q][2] = fmaf(wt, d2, acc[q][2]);
        }
      }
    }
  }
  const float s2 = (float)(K * K);
#pragma unroll
  for (int q = 0; q < 4; ++q)
#pragma unroll
    for (int cc = 0; cc < 3; ++cc) {
      const float m = fmaxf(acc[q][cc] * s2, 0.f) + 1.f;
      h[q][cc] = fmaf(lsc, __logf(m + 1e-7f), h[q][cc]);
    }
}

__global__ __launch_bounds__(256) void level_holder_gate_kernel(
    const float* __restrict__ x, const float* __restrict__ w_down,
    const float* __restrict__ b_down, const float* __restrict__ g1,
    const float* __restrict__ b1, const float* __restrict__ m1,
    const float* __restrict__ v1, const float* __restrict__ anchors,
    const float* __restrict__ memb, const float* __restrict__ wpool,
    const float* __restrict__ bpool, const float* __restrict__ g2,
    const float* __restrict__ b2, const float* __restrict__ m2,
    const float* __restrict__ v2, const float* __restrict__ gk1,
    const float* __restrict__ gk2, const float* __restrict__ gk3,
    const float* __restrict__ gk4, const float* __restrict__ gk5,
    const float* __restrict__ gk6, float* __restrict__ out) {
  __shared__ float sD[EPAD * 3];  // BN+ReLU'd down-projection, haloed + padded

  const int tid = threadIdx.x;
  const int lane = tid & 31;
  const int wave = tid >> 5;
  const int x0 = blockIdx.x * TS;
  const int y0 = blockIdx.y * TS;
  const int bimg = blockIdx.z;
  const float* ximg = x + (size_t)bimg * HW * HW * 3;

  // Interior blocks: every load address of phase 1 (incl. chunk padding rows,
  // ey up to 37 -> gy up to y0+35) lies inside the image -> branch-free path.
  const bool interior =
      (x0 >= 2) && (y0 >= 2) && (x0 + 34 < HW) && (y0 + 35 < HW);
  if (interior)
    down_proj<false>(ximg, w_down, b_down, g1, b1, m1, v1, sD, x0, y0, lane, wave);
  else
    down_proj<true>(ximg, w_down, b_down, g1, b1, m1, v1, sD, x0, y0, lane, wave);
  __syncthreads();

  // ---- phase 2: multiscale log-slope + soft assignment + gate ----
  const int xl = tid & 31;
  const int yl0 = (tid >> 5) * 4;  // 4 consecutive output rows per thread

  float h[4][3];
#pragma unroll
  for (int q = 0; q < 4; ++q)
#pragma unroll
    for (int c = 0; c < 3; ++c) h[q][c] = 0.f;

  accum_scale4<1, 0>(sD, gk1, yl0, xl, LS0, h);
  accum_scale4<2, 0>(sD, gk2, yl0, xl, LS1, h);
  accum_scale4<3, 1>(sD, gk3, yl0, xl, LS2, h);
  accum_scale4<4, 1>(sD, gk4, yl0, xl, LS3, h);
  accum_scale4<5, 2>(sD, gk5, yl0, xl, LS4, h);
  accum_scale4<6, 2>(sD, gk6, yl0, xl, LS5, h);

  float s2v[3], b2v[3], m2v[3];
#pragma unroll
  for (int c = 0; c < 3; ++c) {
    s2v[c] = g2[c] * rsqrtf(v2[c] + 1e-3f);
    b2v[c] = b2[c];
    m2v[c] = m2[c];
  }
  const float bp = bpool[0];
  const size_t imgoff = (size_t)bimg * HW * HW * 3;

#pragma unroll
  for (int q = 0; q < 4; ++q) {
    const int gy = y0 + yl0 + q, gx = x0 + xl;
    const size_t pix = ((size_t)gy * HW + gx) * 3;
    const float* pxc = ximg + pix;
    float* po = out + imgoff + pix;
#pragma unroll
    for (int c = 0; c < 3; ++c) {
      const float hv = h[q][c] * INVS;
      float z[16];
      float zmax = -3.0e38f;
#pragma unroll
      for (int n = 0; n < 16; ++n) {
        const float dlt = hv - anchors[c * 16 + n];  // uniform -> SGPR
        const float zz = -memb[c * 16 + n] * dlt * dlt;
        z[n] = zz;
        zmax = fmaxf(zmax, zz);
      }
      float se = 0.f, sw = 0.f;
#pragma unroll
      for (int n = 0; n < 16; ++n) {
        const float e = __expf(z[n] - zmax);
        se += e;
        sw = fmaf(e, wpool[n], sw);
      }
      const float pooled = sw / se + bp;
      const float t = fmaf(pooled - m2v[c], s2v[c], b2v[c]);
      const float gate = 1.0f / (1.0f + __expf(-t));
      po[c] = pxc[c] * gate;
    }
  }
}

extern "C" void kernel_launch(void* const* d_in, const int* in_sizes, int n_in,
                              void* d_out, int out_size, void* d_ws, size_t ws_size,
                              hipStream_t stream) {
  (void)n_in;
  (void)out_size;
  (void)d_ws;
  (void)ws_size;
  const float* x = (const float*)d_in[0];
  const float* w_down = (const float*)d_in[1];
  const float* b_down = (const float*)d_in[2];
  const float* g1 = (const float*)d_in[3];
  const float* b1 = (const float*)d_in[4];
  const float* m1 = (const float*)d_in[5];
  const float* v1 = (const float*)d_in[6];
  const float* anchors = (const float*)d_in[7];
  const float* memb = (const float*)d_in[8];
  const float* wpool = (const float*)d_in[9];
  const float* bpool = (const float*)d_in[10];
  const float* g2 = (const float*)d_in[11];
  const float* b2 = (const float*)d_in[12];
  const float* m2 = (const float*)d_in[13];
  const float* v2 = (const float*)d_in[14];
  const float* gk1 = (const float*)d_in[15];
  const float* gk2 = (const float*)d_in[16];
  const float* gk3 = (const float*)d_in[17];
  const float* gk4 = (const float*)d_in[18];
  const float* gk5 = (const float*)d_in[19];
  const float* gk6 = (const float*)d_in[20];
  float* out = (float*)d_out;

  const int B = in_sizes[0] / (HW * HW * 3);
  dim3 grid(HW / TS, HW / TS, B);
  dim3 block(256);
  level_holder_gate_kernel<<<grid, block, 0, stream>>>(
      x, w_down, b_down, g1, b1, m1, v1, anchors, memb, wpool, bpool, g2, b2,
      m2, v2, gk1, gk2, gk3, gk4, gk5, gk6, out);
}